// GTCInductive_2233382994256
// MI455X (gfx1250) — compile-verified
//
#include <hip/hip_runtime.h>
#include <math.h>

typedef __attribute__((ext_vector_type(2))) float v2f;
typedef __attribute__((ext_vector_type(8))) float v8f;

#define D_DIM 128
#define J_DIM 64
#define K_TOK 128

__device__ __forceinline__ void atomAddF(float* p, float v) {
  unsafeAtomicAdd(p, v);
}

__global__ void zero_kernel(float4* p, long n4) {
  long i = (long)blockIdx.x * blockDim.x + threadIdx.x;
  long stride = (long)gridDim.x * blockDim.x;
  const float4 z = make_float4(0.f, 0.f, 0.f, 0.f);
  for (; i < n4; i += stride) p[i] = z;
}

__global__ void count_kernel(const int* __restrict__ ei, const int* __restrict__ ek,
                             float* cu, float* cw, int E) {
  int i = blockIdx.x * blockDim.x + threadIdx.x;
  int stride = gridDim.x * blockDim.x;
  for (; i < E; i += stride) {
    atomAddF(&cu[ei[i]], 1.0f);
    atomAddF(&cw[ek[i]], 1.0f);
  }
}

// Fused: e_tile = edge_x[16 edges] @ ep_W + ep_b via f32 WMMA, then scatter
// U_msg[edge_i] += W[edge_k]+e (global atomics, L2-resident) and
// W_msg[edge_k] += U[edge_i]+e (LDS accumulation, flushed once per block).
__global__ __launch_bounds__(256)
void edge_message_kernel(const float* __restrict__ edge_x,
                         const int* __restrict__ edge_i, const int* __restrict__ edge_k,
                         const float* __restrict__ epW, const float* __restrict__ epb,
                         const float* __restrict__ U, const float* __restrict__ W,
                         float* __restrict__ Umsg, float* __restrict__ Wmsg, int E) {
  __shared__ float lW[K_TOK * D_DIM];   // 64 KB local W_msg accumulator
  for (int i = threadIdx.x; i < K_TOK * D_DIM; i += blockDim.x) lW[i] = 0.0f;
  __syncthreads();

  const int lane  = threadIdx.x & 31;
  const int gwave = (blockIdx.x * blockDim.x + threadIdx.x) >> 5;
  const int nwave = (gridDim.x * blockDim.x) >> 5;
  const int mrow  = lane & 15;
  const int khalf = (lane >> 4) << 1;   // 0 or 2 (A/B K-pair per half-wave)
  const int rbase = (lane >> 4) << 3;   // C/D rows: v -> rows v / v+8
  const int ntiles = (E + 15) >> 4;

  for (int tile = gwave; tile < ntiles; tile += nwave) {
    const int e0 = tile << 4;
    int arow = e0 + mrow; if (arow >= E) arow = E - 1;
    const float* ax = edge_x + (long)arow * J_DIM;

    // prefetch next grid-stride tile's A rows (HBM stream) -> global_prefetch_b8
    {
      const int nt2 = tile + nwave;
      if (nt2 < ntiles) {
        int prow = (nt2 << 4) + mrow; if (prow >= E) prow = E - 1;
        const char* pp = (const char*)(edge_x + (long)prow * J_DIM);
        __builtin_prefetch(pp, 0, 3);
        __builtin_prefetch(pp + 128, 0, 3);
      }
    }

    v8f acc[8] = {};
    for (int kb = 0; kb < J_DIM; kb += 4) {
      v2f a;
      a.x = ax[kb + khalf + 0];
      a.y = ax[kb + khalf + 1];
#pragma unroll
      for (int nt = 0; nt < 8; ++nt) {
        const int n = (nt << 4) + mrow;
        v2f b;
        b.x = epW[(kb + khalf + 0) * D_DIM + n];
        b.y = epW[(kb + khalf + 1) * D_DIM + n];
        acc[nt] = __builtin_amdgcn_wmma_f32_16x16x4_f32(
            false, a, false, b, (short)0, acc[nt], false, false);
      }
    }

    float biasr[8];
#pragma unroll
    for (int nt = 0; nt < 8; ++nt) biasr[nt] = epb[(nt << 4) + mrow];

#pragma unroll
    for (int v = 0; v < 8; ++v) {
      const int r = e0 + rbase + v;
      if (r < E) {
        const int ei = edge_i[r];
        const int ek = edge_k[r];
#pragma unroll
        for (int nt = 0; nt < 8; ++nt) {
          const int col = (nt << 4) + mrow;
          const float ev = acc[nt][v] + biasr[nt];
          atomAddF(&Umsg[(long)ei * D_DIM + col], W[ek * D_DIM + col] + ev);
          atomAddF(&lW[ek * D_DIM + col], U[(long)ei * D_DIM + col] + ev);
        }
      }
    }
  }
  __syncthreads();
  for (int i = threadIdx.x; i < K_TOK * D_DIM; i += blockDim.x) {
    float v = lW[i];
    if (v != 0.0f) atomAddF(&Wmsg[i], v);
  }
}

// U_next = relu([U, U_msg/cnt] @ uW[l] + ub[l]); writes in place over the
// U_msg buffer (each wave owns disjoint 16-row tiles; reads precede writes).
__global__ __launch_bounds__(256)
void u_update_kernel(const float* __restrict__ Ucur, float* __restrict__ Ub,
                     const float* __restrict__ cnt, const float* __restrict__ uWl,
                     const float* __restrict__ ubl, int I) {
  const int lane  = threadIdx.x & 31;
  const int gwave = (blockIdx.x * blockDim.x + threadIdx.x) >> 5;
  const int nwave = (gridDim.x * blockDim.x) >> 5;
  const int mrow  = lane & 15;
  const int khalf = (lane >> 4) << 1;
  const int rbase = (lane >> 4) << 3;
  const int ntiles = (I + 15) >> 4;

  for (int tile = gwave; tile < ntiles; tile += nwave) {
    const int r0 = tile << 4;
    int arow = r0 + mrow; if (arow >= I) arow = I - 1;
    const float* ua = Ucur + (long)arow * D_DIM;
    const float* ma = Ub + (long)arow * D_DIM;
    const float rc = 1.0f / fmaxf(cnt[arow], 1.0f);

    v8f acc[8] = {};
    for (int kb = 0; kb < 2 * D_DIM; kb += 4) {
      const int k0 = kb + khalf;     // even; k0 and k0+1 on same side of D
      v2f a;
      if (k0 < D_DIM) { a.x = ua[k0];              a.y = ua[k0 + 1]; }
      else            { a.x = ma[k0 - D_DIM] * rc; a.y = ma[k0 + 1 - D_DIM] * rc; }
#pragma unroll
      for (int nt = 0; nt < 8; ++nt) {
        const int n = (nt << 4) + mrow;
        v2f b;
        b.x = uWl[(k0 + 0) * D_DIM + n];
        b.y = uWl[(k0 + 1) * D_DIM + n];
        acc[nt] = __builtin_amdgcn_wmma_f32_16x16x4_f32(
            false, a, false, b, (short)0, acc[nt], false, false);
      }
    }
#pragma unroll
    for (int nt = 0; nt < 8; ++nt) {
      const int col = (nt << 4) + mrow;
      const float bv = ubl[col];
#pragma unroll
      for (int v = 0; v < 8; ++v) {
        const int r = r0 + rbase + v;
        if (r < I) Ub[(long)r * D_DIM + col] = fmaxf(acc[nt][v] + bv, 0.0f);
      }
    }
  }
}

// W_next = relu([W, W_msg/cnt] @ wW[l] + wb[l])  (tiny: 128x256 @ 256x128)
__global__ void w_update_kernel(const float* __restrict__ Wcur, const float* __restrict__ Wmsg,
                                const float* __restrict__ cnt, const float* __restrict__ wWl,
                                const float* __restrict__ wbl, float* __restrict__ Wnext) {
  const int row = blockIdx.x;
  const int col = threadIdx.x;
  __shared__ float sA[2 * D_DIM];
  const float rc = 1.0f / fmaxf(cnt[row], 1.0f);
  for (int t = threadIdx.x; t < D_DIM; t += blockDim.x) {
    sA[t]         = Wcur[row * D_DIM + t];
    sA[D_DIM + t] = Wmsg[row * D_DIM + t] * rc;
  }
  __syncthreads();
  float acc = wbl[col];
  for (int t = 0; t < 2 * D_DIM; ++t) acc += sA[t] * wWl[t * D_DIM + col];
  Wnext[row * D_DIM + col] = fmaxf(acc, 0.0f);
}

__global__ void add_kernel(const float* a, const float* b, float* c, int n) {
  int i = blockIdx.x * blockDim.x + threadIdx.x;
  if (i < n) c[i] = a[i] + b[i];
}

// C[row,:] = op(A[row,:] @ B + bias); one block per row, one thread per col.
__global__ void gemm_small_kernel(const float* __restrict__ A, const float* __restrict__ B,
                                  const float* __restrict__ bias, float* __restrict__ C,
                                  int Kd, int N, int doRelu) {
  extern __shared__ float sA[];
  const int row = blockIdx.x;
  for (int t = threadIdx.x; t < Kd; t += blockDim.x) sA[t] = A[row * Kd + t];
  __syncthreads();
  const int col = threadIdx.x;
  if (col < N) {
    float acc = bias ? bias[col] : 0.0f;
    for (int t = 0; t < Kd; ++t) acc += sA[t] * B[t * N + col];
    if (doRelu) acc = fmaxf(acc, 0.0f);
    C[row * N + col] = acc;
  }
}

// MHA over K=128 tokens, H=2 heads, dh=64. One block per query token.
__global__ void attention_kernel(const float* __restrict__ q, const float* __restrict__ k_,
                                 const float* __restrict__ v, float* __restrict__ out) {
  const int row = blockIdx.x;
  const int tid = threadIdx.x;           // 128
  __shared__ float sp[K_TOK];
  __shared__ float red[K_TOK];
  for (int h = 0; h < 2; ++h) {
    const int hb = h * 64;
    float sc = 0.0f;
    for (int d = 0; d < 64; ++d)
      sc += q[row * D_DIM + hb + d] * k_[tid * D_DIM + hb + d];
    sc *= 0.125f;                        // 1/sqrt(64)
    sp[tid] = sc; red[tid] = sc;
    __syncthreads();
    for (int s = 64; s > 0; s >>= 1) { if (tid < s) red[tid] = fmaxf(red[tid], red[tid + s]); __syncthreads(); }
    const float mx = red[0];
    __syncthreads();
    const float ex = expf(sp[tid] - mx);
    sp[tid] = ex; red[tid] = ex;
    __syncthreads();
    for (int s = 64; s > 0; s >>= 1) { if (tid < s) red[tid] += red[tid + s]; __syncthreads(); }
    const float inv = 1.0f / red[0];
    __syncthreads();
    if (tid < 64) {
      float acc = 0.0f;
      for (int key = 0; key < K_TOK; ++key) acc += sp[key] * v[key * D_DIM + hb + tid];
      out[row * D_DIM + hb + tid] = acc * inv;
    }
    __syncthreads();
  }
}

__global__ void addln_kernel(const float* __restrict__ a, const float* __restrict__ b,
                             const float* __restrict__ g, const float* __restrict__ be,
                             float* __restrict__ out) {
  const int row = blockIdx.x;
  const int tid = threadIdx.x;           // 128
  __shared__ float red[D_DIM];
  const float x = a[row * D_DIM + tid] + b[row * D_DIM + tid];
  red[tid] = x; __syncthreads();
  for (int s = 64; s > 0; s >>= 1) { if (tid < s) red[tid] += red[tid + s]; __syncthreads(); }
  const float m = red[0] * (1.0f / D_DIM);
  __syncthreads();
  const float d = x - m;
  red[tid] = d * d; __syncthreads();
  for (int s = 64; s > 0; s >>= 1) { if (tid < s) red[tid] += red[tid + s]; __syncthreads(); }
  const float var = red[0] * (1.0f / D_DIM);
  out[row * D_DIM + tid] = d * rsqrtf(var + 1e-5f) * g[tid] + be[tid];
}

// One wave per triple: float4/lane, wave32 shuffle reduce, sigmoid.
__global__ __launch_bounds__(256)
void decode_kernel(const float* __restrict__ U, const float* __restrict__ Wf,
                   const float* __restrict__ Vm, const int* __restrict__ ii,
                   const int* __restrict__ jj, const int* __restrict__ kk,
                   const float* __restrict__ bias, float* __restrict__ out, int NT) {
  const int lane = threadIdx.x & 31;
  int t = (blockIdx.x * blockDim.x + threadIdx.x) >> 5;
  const int nwave = (gridDim.x * blockDim.x) >> 5;
  const float b0 = bias[0];
  for (; t < NT; t += nwave) {
    const int ri = ii[t], rj = jj[t], rk = kk[t];
    // prefetch next iteration's gathered rows while we compute this one
    const int tn = t + nwave;
    if (tn < NT) {
      __builtin_prefetch((const char*)(U + (long)ii[tn] * D_DIM) + (lane & 1) * 256, 0, 3);
      __builtin_prefetch((const char*)(Wf + kk[tn] * D_DIM), 0, 3);
    }
    const float4 ua = ((const float4*)(U  + (long)ri * D_DIM))[lane];
    const float4 wa = ((const float4*)(Wf + rk * D_DIM))[lane];
    const float4 va = ((const float4*)(Vm + rj * D_DIM))[lane];
    float p = ua.x * wa.x * va.x + ua.y * wa.y * va.y +
              ua.z * wa.z * va.z + ua.w * wa.w * va.w;
#pragma unroll
    for (int o = 16; o > 0; o >>= 1) p += __shfl_xor(p, o, 32);
    if (lane == 0) out[t] = 1.0f / (1.0f + expf(-(p + b0)));
  }
}

extern "C" void kernel_launch(void* const* d_in, const int* in_sizes, int n_in,
                              void* d_out, int out_size, void* d_ws, size_t ws_size,
                              hipStream_t stream) {
  const float* edge_x = (const float*)d_in[0];
  const int*   edge_i = (const int*)d_in[1];
  const int*   edge_k = (const int*)d_in[2];
  const int*   idx_i  = (const int*)d_in[3];
  const int*   idx_j  = (const int*)d_in[4];
  const int*   idx_k  = (const int*)d_in[5];
  const float* u0  = (const float*)d_in[6];
  const float* w0  = (const float*)d_in[7];
  const float* epW = (const float*)d_in[8];
  const float* epb = (const float*)d_in[9];
  const float* uW  = (const float*)d_in[10];
  const float* ub  = (const float*)d_in[11];
  const float* wW  = (const float*)d_in[12];
  const float* wb  = (const float*)d_in[13];
  const float* pos = (const float*)d_in[14];
  const float* Wq  = (const float*)d_in[15];
  const float* Wk  = (const float*)d_in[16];
  const float* Wv  = (const float*)d_in[17];
  const float* bq  = (const float*)d_in[18];
  const float* bk  = (const float*)d_in[19];
  const float* bv  = (const float*)d_in[20];
  const float* Wo  = (const float*)d_in[21];
  const float* bo  = (const float*)d_in[22];
  const float* ln1g = (const float*)d_in[23];
  const float* ln1b = (const float*)d_in[24];
  const float* fW1  = (const float*)d_in[25];
  const float* fb1  = (const float*)d_in[26];
  const float* fW2  = (const float*)d_in[27];
  const float* fb2  = (const float*)d_in[28];
  const float* ln2g = (const float*)d_in[29];
  const float* ln2b = (const float*)d_in[30];
  const float* Vm   = (const float*)d_in[31];
  const float* bias = (const float*)d_in[32];

  const int E  = in_sizes[1];
  const int NT = in_sizes[3];
  const int I  = in_sizes[6] / D_DIM;
  const int K  = in_sizes[7] / D_DIM;
  const int L  = in_sizes[11] / D_DIM;

  char* wsp = (char*)d_ws;
  size_t off = 0;
  auto alloc = [&](size_t bytes) -> float* {
    float* p = (float*)(wsp + off);
    off += (bytes + 255) & ~(size_t)255;
    return p;
  };
  float* bufA = alloc((size_t)I * D_DIM * 4);
  float* bufB = alloc((size_t)I * D_DIM * 4);
  float* wmsg = alloc((size_t)K * D_DIM * 4);
  float* wnA  = alloc((size_t)K * D_DIM * 4);
  float* wnB  = alloc((size_t)K * D_DIM * 4);
  float* cu   = alloc((size_t)I * 4);
  float* cw   = alloc((size_t)K * 4);
  float* Xb   = alloc((size_t)K * D_DIM * 4);
  float* qb   = alloc((size_t)K * D_DIM * 4);
  float* kb_  = alloc((size_t)K * D_DIM * 4);
  float* vb   = alloc((size_t)K * D_DIM * 4);
  float* ab   = alloc((size_t)K * D_DIM * 4);
  float* pb   = alloc((size_t)K * D_DIM * 4);
  float* h1   = alloc((size_t)K * D_DIM * 4);
  float* ff   = alloc((size_t)K * 4 * D_DIM * 4);
  float* h2   = alloc((size_t)K * D_DIM * 4);
  float* wfb  = alloc((size_t)K * D_DIM * 4);

  // degree counts (I and K are multiples of 4; D_DIM rows are multiples of 4)
  zero_kernel<<<256, 256, 0, stream>>>((float4*)cu, (long)I / 4);
  zero_kernel<<<1, 256, 0, stream>>>((float4*)cw, (long)K / 4);
  count_kernel<<<2048, 256, 0, stream>>>(edge_i, edge_k, cu, cw, E);

  // message-passing layers (ping-pong U buffers; U_msg buffer becomes U_next)
  const float* Ucur = u0;
  const float* Wcur = w0;
  float* ubufs[2] = { bufB, bufA };
  float* wbufs[2] = { wnA, wnB };
  for (int l = 0; l < L; ++l) {
    float* Umsg  = ubufs[l & 1];
    float* Wnext = wbufs[l & 1];
    zero_kernel<<<2048, 256, 0, stream>>>((float4*)Umsg, (long)I * D_DIM / 4);
    zero_kernel<<<16, 256, 0, stream>>>((float4*)wmsg, (long)K * D_DIM / 4);
    edge_message_kernel<<<512, 256, 0, stream>>>(edge_x, edge_i, edge_k, epW, epb,
                                                 Ucur, Wcur, Umsg, wmsg, E);
    w_update_kernel<<<K, D_DIM, 0, stream>>>(Wcur, wmsg, cw,
                                             wW + (size_t)l * 2 * D_DIM * D_DIM,
                                             wb + (size_t)l * D_DIM, Wnext);
    u_update_kernel<<<512, 256, 0, stream>>>(Ucur, Umsg, cu,
                                             uW + (size_t)l * 2 * D_DIM * D_DIM,
                                             ub + (size_t)l * D_DIM, I);
    Ucur = Umsg; Wcur = Wnext;
  }

  // TimeRefine transformer block (all tiny)
  add_kernel<<<(K * D_DIM + 255) / 256, 256, 0, stream>>>(Wcur, pos, Xb, K * D_DIM);
  gemm_small_kernel<<<K, D_DIM, D_DIM * 4, stream>>>(Xb, Wq, bq, qb, D_DIM, D_DIM, 0);
  gemm_small_kernel<<<K, D_DIM, D_DIM * 4, stream>>>(Xb, Wk, bk, kb_, D_DIM, D_DIM, 0);
  gemm_small_kernel<<<K, D_DIM, D_DIM * 4, stream>>>(Xb, Wv, bv, vb, D_DIM, D_DIM, 0);
  attention_kernel<<<K, D_DIM, 0, stream>>>(qb, kb_, vb, ab);
  gemm_small_kernel<<<K, D_DIM, D_DIM * 4, stream>>>(ab, Wo, bo, pb, D_DIM, D_DIM, 0);
  addln_kernel<<<K, D_DIM, 0, stream>>>(Xb, pb, ln1g, ln1b, h1);
  gemm_small_kernel<<<K, 4 * D_DIM, D_DIM * 4, stream>>>(h1, fW1, fb1, ff, D_DIM, 4 * D_DIM, 1);
  gemm_small_kernel<<<K, D_DIM, 4 * D_DIM * 4, stream>>>(ff, fW2, fb2, h2, 4 * D_DIM, D_DIM, 0);
  addln_kernel<<<K, D_DIM, 0, stream>>>(h1, h2, ln2g, ln2b, wfb);

  // decode triples
  decode_kernel<<<2048, 256, 0, stream>>>(Ucur, wfb, Vm, idx_i, idx_j, idx_k, bias,
                                          (float*)d_out, NT);
}